// GenerativeUpsample_45586782879852
// MI455X (gfx1250) — compile-verified
//
#include <hip/hip_runtime.h>

// ---------------- problem constants (from reference) ----------------
#define BV   8
#define MV   100000
#define VV   100000
#define NV   (BV * MV)     // 800000 rows
#define CC   128           // feature dim
#define KTH  50000u        // k = M - target_num = 100000 - 50000

typedef float v2f __attribute__((ext_vector_type(2)));
typedef float v8f __attribute__((ext_vector_type(8)));

// order-preserving float->uint transform (monotone for all non-NaN floats)
__device__ __forceinline__ unsigned ordf(float f) {
    unsigned u = __float_as_uint(f);
    return (u & 0x80000000u) ? ~u : (u | 0x80000000u);
}

// ======================================================================
// Kernel 1: h = relu(fea @ W1 + b1) -> out_h ; pred = h @ W2 + b2 -> out_pred
//           seg[voxel] = max(seg[voxel], ordf(pred))   (ordered-uint atomicMax)
// One wave computes a 16x128 tile of h via V_WMMA_F32_16X16X4_F32.
// Block = 256 threads = 8 waves -> 128 rows per block. W1 staged in LDS.
// ======================================================================
__global__ __launch_bounds__(256)
void mlp_wmma_kernel(const float* __restrict__ fea,
                     const float* __restrict__ W1,
                     const float* __restrict__ b1,
                     const float* __restrict__ W2,
                     const float* __restrict__ b2,
                     const int*   __restrict__ voxel_ids,
                     float*       __restrict__ out_h,
                     float*       __restrict__ out_pred,
                     unsigned*    __restrict__ seg)
{
    __shared__ float W1s[CC * CC]; // 64 KB: W1 row-major [k][n]

    const int tid = threadIdx.x;
    // cooperative load of W1 into LDS, float4 vectorized (4096 float4 / 256 thr)
    {
        const float4* src = (const float4*)W1;
        float4*       dst = (float4*)W1s;
        #pragma unroll
        for (int i = 0; i < (CC * CC / 4) / 256; ++i)
            dst[tid + i * 256] = src[tid + i * 256];
    }
    __syncthreads();

    const int wave = tid >> 5;
    const int lane = tid & 31;
    const int half = lane >> 4;   // 0: lanes 0-15, 1: lanes 16-31
    const int l16  = lane & 15;
    const int row_base = blockIdx.x * 128 + wave * 16;

    // C/D layout: acc[t][r] = h[m = r + 8*half][n = t*16 + l16]
    // init accumulators with bias b1[n] (same for every row)
    v8f acc[8];
    #pragma unroll
    for (int t = 0; t < 8; ++t) {
        float bn = b1[t * 16 + l16];
        #pragma unroll
        for (int r = 0; r < 8; ++r) acc[t][r] = bn;
    }

    // A layout (16x4 f32): lane l<16 holds M=l, K = k0,k0+1 ; lane l>=16 holds M=l-16, K = k0+2,k0+3
    const float* arow = fea + (size_t)(row_base + l16) * CC + half * 2;

    #pragma unroll 2
    for (int k0 = 0; k0 < CC; k0 += 4) {
        v2f a = *(const v2f*)(arow + k0);   // 8-byte aligned global_load_b64
        #pragma unroll
        for (int t = 0; t < 8; ++t) {
            // B layout (4x16 f32): VGPR0 = K (k0 | k0+2) , VGPR1 = K (k0+1 | k0+3), N = l16
            v2f bf;
            bf[0] = W1s[(k0 + 2 * half + 0) * CC + t * 16 + l16];
            bf[1] = W1s[(k0 + 2 * half + 1) * CC + t * 16 + l16];
            acc[t] = __builtin_amdgcn_wmma_f32_16x16x4_f32(
                false, a, false, bf, (short)0, acc[t], false, false);
        }
    }

    // ReLU
    #pragma unroll
    for (int t = 0; t < 8; ++t)
        #pragma unroll
        for (int r = 0; r < 8; ++r)
            acc[t][r] = fmaxf(acc[t][r], 0.0f);

    // store h (coalesced 64B per half-wave per row)
    #pragma unroll
    for (int t = 0; t < 8; ++t) {
        #pragma unroll
        for (int r = 0; r < 8; ++r) {
            int m = row_base + r + 8 * half;
            out_h[(size_t)m * CC + t * 16 + l16] = acc[t][r];
        }
    }

    // pred[m] = sum_n h[m][n] * W2[n] + b2 ; reduce across the 16 lanes of each half
    float w2l[8];
    #pragma unroll
    for (int t = 0; t < 8; ++t) w2l[t] = W2[t * 16 + l16];
    const float bias2 = b2[0];

    #pragma unroll
    for (int r = 0; r < 8; ++r) {
        float s = 0.0f;
        #pragma unroll
        for (int t = 0; t < 8; ++t) s += acc[t][r] * w2l[t];
        s += __shfl_xor(s, 1, 32);
        s += __shfl_xor(s, 2, 32);
        s += __shfl_xor(s, 4, 32);
        s += __shfl_xor(s, 8, 32);
        if (l16 == 0) {
            int m = row_base + r + 8 * half;
            float pv = s + bias2;
            out_pred[m] = pv;
            atomicMax(&seg[voxel_ids[m]], ordf(pv));
        }
    }
}

// ======================================================================
// Radix-select support kernels (exact k-th smallest per batch)
// ======================================================================
__global__ void init_select_kernel(unsigned* prefix, unsigned* rank)
{
    int t = threadIdx.x;
    if (t < BV) { prefix[t] = 0u; rank[t] = KTH; }
}

// histogram of one byte (pass = 3..0, MSB first) of the ordered key,
// restricted to elements matching the prefix found so far
__global__ __launch_bounds__(256)
void hist_kernel(const float* __restrict__ pred,
                 const int*   __restrict__ voxel_ids,
                 const unsigned* __restrict__ seg,
                 const unsigned* __restrict__ prefix,
                 unsigned*    __restrict__ hist,  // [BV][256]
                 int pass)
{
    int i = blockIdx.x * 256 + threadIdx.x;
    if (i >= NV) return;
    float p = pred[i];
    unsigned ou = ordf(p);
    unsigned lm = seg[voxel_ids[i]];
    // local maxima are masked to +inf (ordered: 0xFF800000)
    unsigned val = (ou == lm) ? 0xFF800000u : ou;
    int b = i / MV;
    int sh = pass * 8;
    if (pass < 3) {
        if ((val >> (sh + 8)) != prefix[b]) return;
    }
    atomicAdd(&hist[b * 256 + ((val >> sh) & 255u)], 1u);
}

// per-batch: scan 256 bins, pick the bin containing rank r, update prefix/rank
__global__ __launch_bounds__(256)
void select_kernel(const unsigned* __restrict__ hist,
                   unsigned* __restrict__ prefix,
                   unsigned* __restrict__ rank,
                   unsigned* __restrict__ thr,
                   int pass)
{
    __shared__ unsigned cum[256];
    int b = blockIdx.x;
    int t = threadIdx.x;
    cum[t] = hist[b * 256 + t];
    __syncthreads();
    // Hillis-Steele inclusive scan
    for (int off = 1; off < 256; off <<= 1) {
        unsigned v = (t >= off) ? cum[t - off] : 0u;
        __syncthreads();
        cum[t] += v;
        __syncthreads();
    }
    unsigned r = rank[b];
    unsigned before = (t > 0) ? cum[t - 1] : 0u;
    if (cum[t] >= r && before < r) {   // exactly one winner
        unsigned np = (prefix[b] << 8) | (unsigned)t;
        prefix[b] = np;
        rank[b]   = r - before;
        if (pass == 0) thr[b] = np;    // full 32-bit ordered threshold
    }
}

// ======================================================================
// keep_target output (third tuple member), as 0.0/1.0 floats
// ======================================================================
__global__ void kt_zero_kernel(float* kt)
{
    int i = blockIdx.x * 256 + threadIdx.x;
    if (i < NV) kt[i] = 0.0f;
}

__global__ void kt_scatter_kernel(const int* __restrict__ tidx, float* kt, int n)
{
    int i = blockIdx.x * 256 + threadIdx.x;
    if (i < n) kt[tidx[i]] = 1.0f;
}

// ======================================================================
// Finalize: keep = is_local_max | (p > thr[b]) | keep_target ;
// one wave per row, zero the 512B row of h when dropped (coalesced float4)
// ======================================================================
__global__ __launch_bounds__(256)
void finalize_kernel(const float* __restrict__ pred,
                     const int*   __restrict__ voxel_ids,
                     const unsigned* __restrict__ seg,
                     const unsigned* __restrict__ thr,
                     const float* __restrict__ kt,
                     float*       __restrict__ out_h)
{
    int gw   = (blockIdx.x * 256 + threadIdx.x) >> 5;  // one wave per row
    int lane = threadIdx.x & 31;
    if (gw >= NV) return;
    float p = pred[gw];
    unsigned ou = ordf(p);
    bool is_lm = (ou == seg[voxel_ids[gw]]);
    int b = gw / MV;
    bool keep = is_lm || (ou > thr[b]) || (kt[gw] != 0.0f);
    if (!keep) {
        float4 z = make_float4(0.f, 0.f, 0.f, 0.f);
        float4* row = (float4*)(out_h + (size_t)gw * CC);
        row[lane] = z;   // 32 lanes x 16B = 512B row
    }
}

// ======================================================================
extern "C" void kernel_launch(void* const* d_in, const int* in_sizes, int n_in,
                              void* d_out, int out_size, void* d_ws, size_t ws_size,
                              hipStream_t stream)
{
    (void)in_sizes; (void)n_in; (void)out_size; (void)ws_size;

    const float* fea        = (const float*)d_in[0];
    const float* W1         = (const float*)d_in[1];
    const float* b1         = (const float*)d_in[2];
    const float* W2         = (const float*)d_in[3];
    const float* b2         = (const float*)d_in[4];
    const int*   voxel_ids  = (const int*)d_in[5];
    const int*   target_idx = (const int*)d_in[6];
    // d_in[7] = target_num (scalar, fixed = M/2 -> KTH hardcoded)

    // output tuple, concatenated flat: out (N*128), pred (N), keep_target (N)
    float* out_h    = (float*)d_out;
    float* out_pred = out_h + (size_t)NV * CC;
    float* out_kt   = out_pred + NV;

    // workspace layout (all u32)
    unsigned* seg    = (unsigned*)d_ws;          // V   ordered segment max
    unsigned* hist   = seg + VV;                 // 4 passes * BV * 256
    unsigned* prefix = hist + 4 * BV * 256;      // BV
    unsigned* rank   = prefix + BV;              // BV
    unsigned* thr    = rank + BV;                // BV

    // zero seg + all histograms (graph-capture safe)
    hipMemsetAsync(d_ws, 0, (size_t)(VV + 4 * BV * 256) * sizeof(unsigned), stream);
    init_select_kernel<<<1, 32, 0, stream>>>(prefix, rank);

    // MLP with f32 WMMA: 6250 blocks x 256 threads (8 waves x 16 rows)
    mlp_wmma_kernel<<<NV / 128, 256, 0, stream>>>(
        fea, W1, b1, W2, b2, voxel_ids, out_h, out_pred, seg);

    // keep_target output
    kt_zero_kernel<<<(NV + 255) / 256, 256, 0, stream>>>(out_kt);
    kt_scatter_kernel<<<(NV / 2 + 255) / 256, 256, 0, stream>>>(target_idx, out_kt, NV / 2);

    // 4-round radix select of the k-th smallest masked value per batch
    for (int pass = 3; pass >= 0; --pass) {
        unsigned* h = hist + (size_t)(3 - pass) * BV * 256;
        hist_kernel<<<(NV + 255) / 256, 256, 0, stream>>>(
            out_pred, voxel_ids, seg, prefix, h, pass);
        select_kernel<<<BV, 256, 0, stream>>>(h, prefix, rank, thr, pass);
    }

    // apply keep mask to h rows
    finalize_kernel<<<(NV * 32 + 255) / 256, 256, 0, stream>>>(
        out_pred, voxel_ids, seg, thr, out_kt, out_h);
}